// Conv4D_42932493091521
// MI455X (gfx1250) — compile-verified
//
#include <hip/hip_runtime.h>

typedef __attribute__((ext_vector_type(2))) float v2f;
typedef __attribute__((ext_vector_type(8))) float v8f;

// x:    (B=2, C=8, U=7, V=7, H=48, W=48) fp32
// conv: (OC=8, K=648), K = (((i0*3+i1)*3+i2)*3+i3)*8 + c
// out:  (B=2, OC=8, U=7, V=7, H=48, W=48) fp32
//
// Implicit GEMM via V_WMMA_F32_16X16X4_F32.
// Block = 96 threads = 3 waves; blockIdx.x encodes (b,u,v,h) -> all row
// addressing is block-uniform (SGPR saddr-form loads). Wave i handles the
// 16-pixel w-tile [16*i, 16*i+16). 162 WMMAs per wave (K=648, 4 per step).
//
//   M rows 0..7 = output channels (rows 8..15 duplicate rows 0..7 via m&7,
//                 garbage confined to unread D rows; keeps EXEC all-1s)
//   N cols      = 16 consecutive w pixels

#define S_H 48
#define S_V 2304       // 48*48
#define S_U 16128      // 7*2304
#define S_C 112896     // 7*16128
#define S_B 903168     // 8*112896

__global__ __launch_bounds__(96) void conv4d_wmma_f32(
    const float* __restrict__ x,
    const float* __restrict__ conv,
    const float* __restrict__ bias,
    float* __restrict__ out)
{
    // ---- block-uniform coordinates (stay in SGPRs)
    int rid = blockIdx.x;
    const int h = rid % 48; rid /= 48;
    const int v = rid % 7;  rid /= 7;
    const int u = rid % 7;  rid /= 7;
    const int b = rid;

    const int lane = threadIdx.x & 31;
    const int w0   = ((threadIdx.x >> 5) & 3) << 4;   // wave id * 16

    const int n    = lane & 15;          // N column
    const int koff = (lane >> 4) << 1;   // K sub-offset in group of 4: 0 or 2

    // ---- per-lane loop-invariant B byte-offsets / w-validity for the
    // 24-long (i3,c) K-block: j = kq*4 + koff + p ; i3 = j>>3 ; c = j&7
    int  po[12];
    bool wv[12];
    #pragma unroll
    for (int kq = 0; kq < 6; ++kq) {
        #pragma unroll
        for (int p = 0; p < 2; ++p) {
            const int j  = kq * 4 + koff + p;
            const int i3 = j >> 3;
            const int c  = j & 7;
            const int wc = w0 + n + i3 - 1;              // w coordinate
            po[kq * 2 + p] = (c * S_C + wc) * 4;         // bytes from row base
            wv[kq * 2 + p] = ((unsigned)wc < 48u);
        }
    }

    // ---- per-lane A byte offset (row m clamped with m&7; K sub-offset koff)
    const int abase = ((lane & 7) * 648 + koff) * 4;     // bytes

    // ---- accumulator init with bias (VGPR r <-> M in {r, r+8} -> oc = r)
    v8f acc;
    #pragma unroll
    for (int r = 0; r < 8; ++r) acc[r] = bias[r];

    const char* xc = (const char*)x;

    for (int i0 = 0; i0 < 3; ++i0) {
        const int  uu  = u + i0 - 1;
        const bool uok = ((unsigned)uu < 7u);
        for (int i1 = 0; i1 < 3; ++i1) {
            const int  vv  = v + i1 - 1;
            const bool vok = ((unsigned)vv < 7u);
            for (int i2 = 0; i2 < 3; ++i2) {
                const int  hh  = h + i2 - 1;
                const bool rok = uok && vok && ((unsigned)hh < 48u); // uniform

                // uniform row base (SGPR) + uniform fallback offset -> x[0]
                const long long rowbyte =
                    ((long long)b * S_B + (long long)uu * S_U +
                     (long long)vv * S_V + (long long)hh * S_H) * 4;
                const char* rowp = xc + rowbyte;          // SGPR pair
                const int   fb   = (int)(-rowbyte);       // clamp to x[0]

                // uniform A pointer for this tap
                const int tapbase = ((i0 * 3 + i1) * 3 + i2) * 24;   // *24 K
                const char* ap = (const char*)(conv + tapbase);      // SGPR

                #pragma unroll
                for (int kq = 0; kq < 6; ++kq) {
                    // A fragment: saddr + invariant lane offset + imm (kq*16B)
                    const v2f a = *(const v2f*)(ap + abase + kq * 16);

                    // B fragment: saddr-form b32 loads, 32-bit offset clamp
                    const bool v0 = rok && wv[kq * 2 + 0];
                    const bool v1 = rok && wv[kq * 2 + 1];
                    const int  o0 = v0 ? po[kq * 2 + 0] : fb;
                    const int  o1 = v1 ? po[kq * 2 + 1] : fb;
                    const float t0 = *(const float*)(rowp + o0);
                    const float t1 = *(const float*)(rowp + o1);
                    v2f bf;
                    bf.x = v0 ? t0 : 0.0f;
                    bf.y = v1 ? t1 : 0.0f;

                    acc = __builtin_amdgcn_wmma_f32_16x16x4_f32(
                        false, a, false, bf, (short)0, acc, false, false);
                }
            }
        }
    }

    // ---- store: lanes 0..15 hold pixel n for all 8 channels (VGPR r = oc r).
    // Output strides match input (OC == C == 8). saddr + imm-offset stores.
    if (lane < 16) {
        float* op = out + ((long long)b * S_B + (long long)u * S_U +
                           (long long)v * S_V + (long long)h * S_H);
        const int wn = w0 + n;
        #pragma unroll
        for (int oc = 0; oc < 8; ++oc) {
            op[oc * S_C + wn] = acc[oc];
        }
    }
}

extern "C" void kernel_launch(void* const* d_in, const int* in_sizes, int n_in,
                              void* d_out, int out_size, void* d_ws, size_t ws_size,
                              hipStream_t stream) {
    const float* x    = (const float*)d_in[0];
    const float* conv = (const float*)d_in[1];
    const float* bias = (const float*)d_in[2];
    float* out        = (float*)d_out;

    // blocks over (b,u,v,h) = 2*7*7*48 = 4704 ; 3 waves/block cover w
    const int blocks = 2 * 7 * 7 * 48;
    conv4d_wmma_f32<<<blocks, 96, 0, stream>>>(x, conv, bias, out);
}